// FocalDetectionLoss_40948218200547
// MI455X (gfx1250) — compile-verified
//
#include <hip/hip_runtime.h>
#include <math.h>

// ---------------- problem constants ----------------
static constexpr int BATCH = 64;
static constexpr int NBOX  = 20;
static constexpr int NCLS  = 20;
static constexpr int G0 = 52, G1 = 26, G2 = 13;
static constexpr int CELLS0 = BATCH * G0 * G0 * 3;   // 519168
static constexpr int CELLS1 = BATCH * G1 * G1 * 3;   // 129792
static constexpr int CELLS2 = BATCH * G2 * G2 * 3;   //  32448
static constexpr int T_TOTAL = CELLS0 + CELLS1 + CELLS2;

__constant__ float d_anchors[3][3][2] = {
    {{10.f, 13.f}, {16.f, 30.f}, {33.f, 23.f}},
    {{30.f, 61.f}, {62.f, 45.f}, {59.f, 119.f}},
    {{116.f, 90.f}, {156.f, 198.f}, {373.f, 326.f}}};

// ---------------- WMMA wave32 reduction ----------------
typedef __attribute__((ext_vector_type(2))) float v2f;
typedef __attribute__((ext_vector_type(8))) float v8f;

// Sum one float across all 32 lanes of the wave using
// v_wmma_f32_16x16x4_f32 with B = ones:
//   A[m][k]: lane m (0-15) -> k=0 (vgpr0), k=1 (vgpr1); lane m+16 -> k=2,3
//   D[m][n] = rowsum(m) = x_m + x_{m+16}  (v1 set to 0)
//   Lanes 0-15 hold D rows 0..7 in 8 vgprs, lanes 16-31 rows 8..15.
__device__ __forceinline__ float wave_sum_wmma(float x) {
  v2f a; a[0] = x;    a[1] = 0.0f;
  v2f b; b[0] = 1.0f; b[1] = 1.0f;
  v8f c = {};
  v8f d = __builtin_amdgcn_wmma_f32_16x16x4_f32(
      /*neg_a=*/false, a, /*neg_b=*/false, b,
      /*c_mod=*/(short)0, c, /*reuse_a=*/false, /*reuse_b=*/false);
  float t = d[0] + d[1] + d[2] + d[3] + d[4] + d[5] + d[6] + d[7];
  t += __shfl_xor(t, 16, 32);   // combine rows 0-7 with rows 8-15
  return t;
}

// ---------------- math helpers ----------------
__device__ __forceinline__ float sigmoid_f(float x) {
  return 1.0f / (1.0f + expf(-x));
}

__device__ __forceinline__ float focal_elem(float p, float t) {
  float bce = -(t * logf(fmaxf(p, 1e-12f)) +
                (1.0f - t) * logf(fmaxf(1.0f - p, 1e-12f)));
  float p_t = p * t + (1.0f - p) * (1.0f - t);
  float om  = 1.0f - p_t;
  return 0.25f * om * om * bce;   // ALPHA * (1-p_t)^GAMMA * bce
}

// ---------------- kernels ----------------
__global__ void fdl_init_kernel(int* __restrict__ codes, float* __restrict__ acc,
                                int total) {
  int i = blockIdx.x * blockDim.x + threadIdx.x;
  if (i < 12) acc[i] = 0.0f;
  int stride = gridDim.x * blockDim.x;
  for (int t = i; t < total; t += stride) codes[t] = -1;
}

__global__ void fdl_build_targets_kernel(const float* __restrict__ boxes,
                                         const int* __restrict__ labels,
                                         int* __restrict__ codes,
                                         float* __restrict__ bbox_t) {
  int idx = blockIdx.x * blockDim.x + threadIdx.x;
  if (idx >= 3 * BATCH * NBOX) return;
  int s = idx / (BATCH * NBOX);
  int r = idx % (BATCH * NBOX);
  int b = r / NBOX;
  int n = r % NBOX;

  int G = (s == 0) ? G0 : ((s == 1) ? G1 : G2);
  size_t base = (s == 0) ? 0
              : (s == 1) ? (size_t)CELLS0
                         : (size_t)CELLS0 + (size_t)CELLS1;

  const float* bx = boxes + ((size_t)b * NBOX + n) * 4;
  float sc = (float)G / 416.0f;
  float x1 = bx[0] * sc, y1 = bx[1] * sc, x2 = bx[2] * sc, y2 = bx[3] * sc;
  float cx = 0.5f * (x1 + x2);
  float cy = 0.5f * (y1 + y2);
  int gx = (int)cx;                // truncation == floor for positive coords
  int gy = (int)cy;
  float bw = x2 - x1;
  float bh = y2 - y1;

  float best_iou = -1.0f;
  int best = 0;
  #pragma unroll
  for (int a = 0; a < 3; ++a) {
    float aw = d_anchors[s][a][0] * sc;
    float ah = d_anchors[s][a][1] * sc;
    float inter = fminf(bw, aw) * fminf(bh, ah);
    float uni = bw * bh + aw * ah - inter;
    float iou = inter / (uni + 1e-6f);
    if (iou > best_iou) { best_iou = iou; best = a; }   // first-max wins
  }

  if (gx >= 0 && gx < G && gy >= 0 && gy < G) {
    size_t cell = base + ((((size_t)b * G + gy) * G + gx) * 3 + best);
    codes[cell] = labels[(size_t)b * NBOX + n];
    float* bt = bbox_t + cell * 4;
    bt[0] = cx - (float)gx;
    bt[1] = cy - (float)gy;
    bt[2] = bw;
    bt[3] = bh;
  }
}

// One streaming pass over all cells of a scale: obj focal everywhere,
// IoU + cls focal at positive cells, wave-reduced via WMMA, then atomics.
__global__ void fdl_scale_loss_kernel(const float* __restrict__ pred,
                                      const int* __restrict__ codes,
                                      const float* __restrict__ bbox_t,
                                      float* __restrict__ acc,   // [4]: obj,bbox,cls,cnt
                                      int cells) {
  float obj_acc = 0.0f, bb_acc = 0.0f, cls_acc = 0.0f, cnt_acc = 0.0f;

  int i = blockIdx.x * blockDim.x + threadIdx.x;
  int stride = gridDim.x * blockDim.x;

  for (int t = i; t < cells; t += stride) {
    const float* pc = pred + (size_t)t * 25;
    // stay ahead of the stream
    __builtin_prefetch(pred + (size_t)(t + stride) * 25, 0, 3);

    float p_obj = sigmoid_f(pc[4]);
    int code = codes[t];
    float tm = (code >= 0) ? 1.0f : 0.0f;
    obj_acc += focal_elem(p_obj, tm);

    if (code >= 0) {
      // predicted box (cx,cy,w,h) -> corners
      float px = sigmoid_f(pc[0]);
      float py = sigmoid_f(pc[1]);
      float pw = expf(fminf(pc[2], 10.0f));
      float ph = expf(fminf(pc[3], 10.0f));
      float px1 = px - 0.5f * pw, py1 = py - 0.5f * ph;
      float px2 = px + 0.5f * pw, py2 = py + 0.5f * ph;

      const float* bt = bbox_t + (size_t)t * 4;
      float tx = bt[0], ty = bt[1], tw = bt[2], th = bt[3];
      float tx1 = tx - 0.5f * tw, ty1 = ty - 0.5f * th;
      float tx2 = tx + 0.5f * tw, ty2 = ty + 0.5f * th;

      float ix1 = fmaxf(px1, tx1), iy1 = fmaxf(py1, ty1);
      float ix2 = fminf(px2, tx2), iy2 = fminf(py2, ty2);
      float inter = fmaxf(ix2 - ix1, 0.0f) * fmaxf(iy2 - iy1, 0.0f);
      float pa = (px2 - px1) * (py2 - py1);
      float ta = (tx2 - tx1) * (ty2 - ty1);
      float iou = inter / (pa + ta - inter + 1e-6f);
      bb_acc += 1.0f - iou;
      cnt_acc += 1.0f;

      #pragma unroll
      for (int c = 0; c < NCLS; ++c) {
        float pcls = sigmoid_f(pc[5 + c]);
        float tc = (c == code) ? 1.0f : 0.0f;
        cls_acc += focal_elem(pcls, tc);
      }
    }
  }

  // uniform control flow here -> EXEC all ones -> WMMA reduction is legal
  float obj_s = wave_sum_wmma(obj_acc);
  float bb_s  = wave_sum_wmma(bb_acc);
  float cls_s = wave_sum_wmma(cls_acc);
  float cnt_s = wave_sum_wmma(cnt_acc);

  if ((threadIdx.x & 31) == 0) {
    atomicAdd(&acc[0], obj_s);
    atomicAdd(&acc[1], bb_s);
    atomicAdd(&acc[2], cls_s);
    atomicAdd(&acc[3], cnt_s);
  }
}

__global__ void fdl_finalize_kernel(const float* __restrict__ acc,
                                    float* __restrict__ out) {
  if (blockIdx.x == 0 && threadIdx.x == 0) {
    const float cells[3] = {(float)CELLS0, (float)CELLS1, (float)CELLS2};
    float total = 0.0f;
    #pragma unroll
    for (int s = 0; s < 3; ++s) {
      const float* a = acc + s * 4;
      float obj_loss = a[0] / cells[s];
      float cnt = a[3];
      float denom = fmaxf(cnt, 1.0f);
      float bbox_loss = (cnt > 0.0f) ? a[1] / denom : 0.0f;
      float cls_loss  = (cnt > 0.0f) ? a[2] / (denom * (float)NCLS) : 0.0f;
      total += 1.0f * obj_loss + 5.0f * bbox_loss + 1.0f * cls_loss;
    }
    out[0] = total / 3.0f;
  }
}

// ---------------- launch ----------------
extern "C" void kernel_launch(void* const* d_in, const int* in_sizes, int n_in,
                              void* d_out, int out_size, void* d_ws, size_t ws_size,
                              hipStream_t stream) {
  (void)in_sizes; (void)n_in; (void)out_size; (void)ws_size;

  const float* pred[3] = {(const float*)d_in[0], (const float*)d_in[1],
                          (const float*)d_in[2]};
  const float* boxes = (const float*)d_in[3];
  const int* labels = (const int*)d_in[4];
  float* out = (float*)d_out;

  // workspace layout: [acc: 12 floats (pad to 64B)] [codes: T ints] [bbox: T*4 floats]
  char* ws = (char*)d_ws;
  float* acc = (float*)ws;
  int* codes = (int*)(ws + 64);
  float* bbox = (float*)(ws + 64 + (size_t)T_TOTAL * sizeof(int));

  const int TPB = 256;

  // reset accumulators + clear target codes (every call, deterministic)
  fdl_init_kernel<<<(T_TOTAL + TPB - 1) / TPB, TPB, 0, stream>>>(codes, acc, T_TOTAL);

  // scatter targets (3*64*20 = 3840 boxes)
  fdl_build_targets_kernel<<<(3 * BATCH * NBOX + TPB - 1) / TPB, TPB, 0, stream>>>(
      boxes, labels, codes, bbox);

  const int cells_per_scale[3] = {CELLS0, CELLS1, CELLS2};
  const size_t cell_base[3] = {0, (size_t)CELLS0, (size_t)CELLS0 + CELLS1};
  for (int s = 0; s < 3; ++s) {
    int cells = cells_per_scale[s];
    int blocks = (cells + TPB - 1) / TPB;
    if (blocks > 2048) blocks = 2048;
    fdl_scale_loss_kernel<<<blocks, TPB, 0, stream>>>(
        pred[s], codes + cell_base[s], bbox + cell_base[s] * 4, acc + s * 4, cells);
  }

  fdl_finalize_kernel<<<1, 32, 0, stream>>>(acc, out);
}